// ReverseTModel_45500883534159
// MI455X (gfx1250) — compile-verified
//
#include <hip/hip_runtime.h>
#include <stdint.h>
#include <math.h>

namespace {
constexpr int HIN   = 81;
constexpr int PLANE = HIN * HIN;     // 6561 dwords per channel plane
constexpr int NCH   = 12;            // 3 + 3 + 2 + 4 channel planes
constexpr int WOUT  = 512;
constexpr int HOUT  = 512;
constexpr int TILES = 8;             // row tiles per image
constexpr int ROWS  = HOUT / TILES;  // 64 rows per tile
constexpr int BLOCK = 1024;
}

// Issue per-lane async global->LDS dword copies (CDNA5 ASYNCcnt path).
__device__ __forceinline__ void async_stage(const float* __restrict__ src,
                                            float* lds_dst, int ndw, int tid) {
  for (int i = tid; i < ndw; i += BLOCK) {
    uint32_t ldsb = (uint32_t)(uintptr_t)(lds_dst + i);   // low 32 bits = LDS byte offset
    uint32_t goff = (uint32_t)(i * 4);                    // byte offset from SGPR base
    asm volatile("global_load_async_to_lds_b32 %0, %1, %2"
                 :: "v"(ldsb), "v"(goff), "s"(src) : "memory");
  }
}

// nearest-sample coords for map m: pixel index (border-clamped) + zeros-padding weight
__device__ __forceinline__ void map_coords(const float* R, int m, float X, float Y,
                                           int& pp, float& w) {
  float gx = fmaf(R[6 * m + 0], X, fmaf(R[6 * m + 1], Y, R[6 * m + 2]));
  float gy = fmaf(R[6 * m + 3], X, fmaf(R[6 * m + 4], Y, R[6 * m + 5]));
  float fx = rintf((gx + 1.0f) * 40.0f);   // (gx+1)*0.5*(81-1), round-half-even == jnp.round
  float fy = rintf((gy + 1.0f) * 40.0f);
  bool valid = (fx >= 0.0f) & (fx <= 80.0f) & (fy >= 0.0f) & (fy <= 80.0f);
  float cx = fminf(fmaxf(fx, 0.0f), 80.0f);
  float cy = fminf(fmaxf(fy, 0.0f), 80.0f);
  pp = (int)cy * HIN + (int)cx;
  w  = valid ? 1.0f : 0.0f;
}

__global__ __launch_bounds__(BLOCK, 1)
void reverse_t_fused(const float* __restrict__ p0, const float* __restrict__ p1,
                     const float* __restrict__ p2, const float* __restrict__ p3,
                     const float* __restrict__ theta, float* __restrict__ out) {
  // 12 channel planes of one batch item, fp32, resident in LDS (314,928 B)
  __shared__ float lds[NCH * PLANE];
  __shared__ float rt[4][6];

  const int n    = blockIdx.y;
  const int tile = blockIdx.x;
  const int tid  = threadIdx.x;

  // ---- 1) stage raw pred planes for batch n into LDS (async, ASYNCcnt) ----
  async_stage(p0 + (size_t)n * 3 * PLANE, lds + 0 * PLANE, 3 * PLANE, tid);
  async_stage(p1 + (size_t)n * 3 * PLANE, lds + 3 * PLANE, 3 * PLANE, tid);
  async_stage(p2 + (size_t)n * 2 * PLANE, lds + 6 * PLANE, 2 * PLANE, tid);
  async_stage(p3 + (size_t)n * 4 * PLANE, lds + 8 * PLANE, 4 * PLANE, tid);

  // ---- 2) invert the 4 affines for this n (threads 0..3) ----
  if (tid < 4) {
    const float* t = theta + (size_t)n * 24 + tid * 6;
    float a = t[0], b = t[1], e = t[2];
    float c = t[3], d = t[4], f = t[5];
    float inv = 1.0f / (a * d - b * c);
    rt[tid][0] =  d * inv;  rt[tid][1] = -b * inv;  rt[tid][2] = (b * f - d * e) * inv;
    rt[tid][3] = -c * inv;  rt[tid][4] =  a * inv;  rt[tid][5] = (c * e - a * f) * inv;
  }

  asm volatile("s_wait_asynccnt 0" ::: "memory");
  __syncthreads();

  // ---- 3) channel softmax in LDS (per pixel, per map) ----
  for (int p = tid; p < PLANE; p += BLOCK) {
    { // map0: planes 0..2
      float v0 = lds[0 * PLANE + p], v1 = lds[1 * PLANE + p], v2 = lds[2 * PLANE + p];
      float mx = fmaxf(v0, fmaxf(v1, v2));
      float e0 = expf(v0 - mx), e1 = expf(v1 - mx), e2 = expf(v2 - mx);
      float r = 1.0f / (e0 + e1 + e2);
      lds[0 * PLANE + p] = e0 * r; lds[1 * PLANE + p] = e1 * r; lds[2 * PLANE + p] = e2 * r;
    }
    { // map1: planes 3..5
      float v0 = lds[3 * PLANE + p], v1 = lds[4 * PLANE + p], v2 = lds[5 * PLANE + p];
      float mx = fmaxf(v0, fmaxf(v1, v2));
      float e0 = expf(v0 - mx), e1 = expf(v1 - mx), e2 = expf(v2 - mx);
      float r = 1.0f / (e0 + e1 + e2);
      lds[3 * PLANE + p] = e0 * r; lds[4 * PLANE + p] = e1 * r; lds[5 * PLANE + p] = e2 * r;
    }
    { // map2: planes 6..7
      float v0 = lds[6 * PLANE + p], v1 = lds[7 * PLANE + p];
      float mx = fmaxf(v0, v1);
      float e0 = expf(v0 - mx), e1 = expf(v1 - mx);
      float r = 1.0f / (e0 + e1);
      lds[6 * PLANE + p] = e0 * r; lds[7 * PLANE + p] = e1 * r;
    }
    { // map3: planes 8..11
      float v0 = lds[8 * PLANE + p], v1 = lds[9 * PLANE + p];
      float v2 = lds[10 * PLANE + p], v3 = lds[11 * PLANE + p];
      float mx = fmaxf(fmaxf(v0, v1), fmaxf(v2, v3));
      float e0 = expf(v0 - mx), e1 = expf(v1 - mx), e2 = expf(v2 - mx), e3 = expf(v3 - mx);
      float r = 1.0f / (e0 + e1 + e2 + e3);
      lds[8 * PLANE + p]  = e0 * r; lds[9 * PLANE + p]  = e1 * r;
      lds[10 * PLANE + p] = e2 * r; lds[11 * PLANE + p] = e3 * r;
    }
  }
  __syncthreads();

  // ---- 4) sample the output tile ----
  float R[24];
#pragma unroll
  for (int j = 0; j < 24; ++j) R[j] = (&rt[0][0])[j];

  const float sx = 2.0f / (WOUT - 1);
  const float sy = 2.0f / (HOUT - 1);
  const int y0 = tile * ROWS;
  const size_t CS = (size_t)HOUT * WOUT;           // output channel stride
  float* ob = out + (size_t)n * 9 * CS;

  for (int p = tid; p < ROWS * WOUT; p += BLOCK) {
    int y = y0 + (p >> 9);
    int x = p & (WOUT - 1);
    float X = x * sx - 1.0f;
    float Y = y * sy - 1.0f;

    int pp; float w;
    // map0 (planes 0..2): bg0 = ch0 border; fg0 = ch1*w, fg1 = ch2*w
    map_coords(R, 0, X, Y, pp, w);
    float bg  = lds[0 * PLANE + pp];
    float fg0 = lds[1 * PLANE + pp] * w;
    float fg1 = lds[2 * PLANE + pp] * w;
    // map1 (planes 3..5)
    map_coords(R, 1, X, Y, pp, w);
    bg *= lds[3 * PLANE + pp];
    float fg2 = lds[4 * PLANE + pp] * w;
    float fg3 = lds[5 * PLANE + pp] * w;
    // map2 (planes 6..7)
    map_coords(R, 2, X, Y, pp, w);
    bg *= lds[6 * PLANE + pp];
    float fg4 = lds[7 * PLANE + pp] * w;
    // map3 (planes 8..11)
    map_coords(R, 3, X, Y, pp, w);
    bg *= lds[8 * PLANE + pp];
    float fg5 = lds[9 * PLANE + pp] * w;
    float fg6 = lds[10 * PLANE + pp] * w;
    float fg7 = lds[11 * PLANE + pp] * w;

    size_t o = (size_t)y * WOUT + x;
    // output channel order: [bg, fg0, fg2, fg1, fg3, fg4, fg5, fg6, fg7]
    __builtin_nontemporal_store(bg,  ob + 0 * CS + o);
    __builtin_nontemporal_store(fg0, ob + 1 * CS + o);
    __builtin_nontemporal_store(fg2, ob + 2 * CS + o);
    __builtin_nontemporal_store(fg1, ob + 3 * CS + o);
    __builtin_nontemporal_store(fg3, ob + 4 * CS + o);
    __builtin_nontemporal_store(fg4, ob + 5 * CS + o);
    __builtin_nontemporal_store(fg5, ob + 6 * CS + o);
    __builtin_nontemporal_store(fg6, ob + 7 * CS + o);
    __builtin_nontemporal_store(fg7, ob + 8 * CS + o);
  }
}

extern "C" void kernel_launch(void* const* d_in, const int* in_sizes, int n_in,
                              void* d_out, int out_size, void* d_ws, size_t ws_size,
                              hipStream_t stream) {
  (void)n_in; (void)out_size; (void)d_ws; (void)ws_size;
  const float* p0 = (const float*)d_in[0];
  const float* p1 = (const float*)d_in[1];
  const float* p2 = (const float*)d_in[2];
  const float* p3 = (const float*)d_in[3];
  const float* th = (const float*)d_in[4];
  const int N = in_sizes[4] / 24;  // theta is (N,4,2,3)

  dim3 grid(TILES, N);
  reverse_t_fused<<<grid, BLOCK, 0, stream>>>(p0, p1, p2, p3, th, (float*)d_out);
}